// FCOS_81200651698452
// MI455X (gfx1250) — compile-verified
//
#include <hip/hip_runtime.h>
#include <stdint.h>

typedef __attribute__((ext_vector_type(2))) float v2f;
typedef __attribute__((ext_vector_type(8))) float v8f;
typedef unsigned long long ull;

#define NCLS 80
#define KTOP 1000
#define NTOT 3000
#define NWORDS 94        // ceil(3008/32)
#define CAND_CAP 4096
#define CONF 0.05f
#define NMS_T 0.6f
#define COFF 10000.0f

struct Ws {
  unsigned hist[2048];
  unsigned hist2[2048];
  unsigned selBin;
  unsigned cntAbove;
  unsigned threshBits;
  unsigned candCount;
  unsigned pad0[4];
  ull      cand[CAND_CAP];        // (score_bits<<32) | ~flat_idx
  float    scores[3072];
  int      labels[3072];
  float    boxes[3072 * 4];       // raw (un-offset) boxes
  int      order[3072];           // sorted position -> original index
  unsigned maskmat[3008 * NWORDS];
};

__device__ __forceinline__ float sigmoidf(float x) { return 1.0f / (1.0f + expf(-x)); }

// ---------------- per-level zero ----------------
__global__ void k_zero(Ws* w) {
  int t = blockIdx.x * 256 + threadIdx.x;
  if (t < 2048) w->hist[t] = 0;
  else if (t < 4096) w->hist2[t - 2048] = 0;
  else if (t == 4096) w->candCount = 0;
}

// ---------------- pass 1: coarse histogram of score bits (async-to-LDS staged) ----------------
__global__ __launch_bounds__(256) void k_score_hist(const float* __restrict__ cls,
                                                    const float* __restrict__ ctn,
                                                    Ws* w) {
  __shared__ unsigned lh[2048];
  __shared__ __align__(16) float stage[256 * 4];
  for (int i = threadIdx.x; i < 2048; i += 256) lh[i] = 0;

  int base = (int)(blockIdx.x * 256 + threadIdx.x) * 4;   // 4 consecutive classes, same anchor
  // gfx1250 async copy global->LDS (ASYNCcnt-tracked), 16B per lane
  unsigned lds_off = (unsigned)(uintptr_t)(&stage[threadIdx.x * 4]);
  unsigned gbyte   = (unsigned)base * 4u;
  asm volatile("global_load_async_to_lds_b128 %0, %1, %2"
               :: "v"(lds_off), "v"(gbyte), "s"(cls) : "memory");
  float sct = sigmoidf(ctn[base / NCLS]);
  asm volatile("s_wait_asynccnt 0" ::: "memory");
  __syncthreads();   // lh zeroed; own staged lanes ready

  float4 cv = *(const float4*)(&stage[threadIdx.x * 4]);
  float c[4] = {cv.x, cv.y, cv.z, cv.w};
#pragma unroll
  for (int k = 0; k < 4; ++k) {
    float s = sqrtf(sigmoidf(c[k]) * sct);     // in (0,1): positive -> bits are order-preserving
    unsigned bits = __float_as_uint(s);
    atomicAdd(&lh[bits >> 19], 1u);
  }
  __syncthreads();
  for (int i = threadIdx.x; i < 2048; i += 256)
    if (lh[i]) atomicAdd(&w->hist[i], lh[i]);
}

// ---------------- pick coarse bin B (suffix count) ----------------
__global__ void k_pick1(Ws* w) {
  if (threadIdx.x != 0) return;
  unsigned cum = 0;
  for (int b = 2047; b >= 0; --b) {
    unsigned c = w->hist[b];
    if (cum + c >= KTOP) { w->selBin = (unsigned)b; w->cntAbove = cum; return; }
    cum += c;
  }
  w->selBin = 0; w->cntAbove = 0;
}

// ---------------- pass 2: refine histogram inside bin B ----------------
__global__ __launch_bounds__(256) void k_hist2(const float* __restrict__ cls,
                                               const float* __restrict__ ctn,
                                               Ws* w) {
  __shared__ unsigned lh[2048];
  for (int i = threadIdx.x; i < 2048; i += 256) lh[i] = 0;
  __syncthreads();
  unsigned B = w->selBin;
  int base = (int)(blockIdx.x * 256 + threadIdx.x) * 4;
  float4 cv = *(const float4*)(cls + base);
  float sct = sigmoidf(ctn[base / NCLS]);
  float c[4] = {cv.x, cv.y, cv.z, cv.w};
#pragma unroll
  for (int k = 0; k < 4; ++k) {
    unsigned bits = __float_as_uint(sqrtf(sigmoidf(c[k]) * sct));
    if ((bits >> 19) == B) atomicAdd(&lh[(bits >> 8) & 0x7FF], 1u);
  }
  __syncthreads();
  for (int i = threadIdx.x; i < 2048; i += 256)
    if (lh[i]) atomicAdd(&w->hist2[i], lh[i]);
}

__global__ void k_pick2(Ws* w) {
  if (threadIdx.x != 0) return;
  unsigned target = KTOP - w->cntAbove;   // >= 1 by construction
  unsigned cum = 0;
  for (int b = 2047; b >= 0; --b) {
    unsigned c = w->hist2[b];
    if (cum + c >= target) { w->threshBits = (w->selBin << 19) | ((unsigned)b << 8); return; }
    cum += c;
  }
  w->threshBits = w->selBin << 19;
}

// ---------------- pass 3: compact candidates >= threshold ----------------
__global__ __launch_bounds__(256) void k_compact(const float* __restrict__ cls,
                                                 const float* __restrict__ ctn,
                                                 Ws* w) {
  unsigned tb = w->threshBits;
  int base = (int)(blockIdx.x * 256 + threadIdx.x) * 4;
  float4 cv = *(const float4*)(cls + base);
  float sct = sigmoidf(ctn[base / NCLS]);
  float c[4] = {cv.x, cv.y, cv.z, cv.w};
#pragma unroll
  for (int k = 0; k < 4; ++k) {
    unsigned bits = __float_as_uint(sqrtf(sigmoidf(c[k]) * sct));
    if (bits >= tb) {
      unsigned pos = atomicAdd(&w->candCount, 1u);
      if (pos < CAND_CAP)
        w->cand[pos] = ((ull)bits << 32) | (unsigned)(~(unsigned)(base + k));
    }
  }
}

// ---------------- bitonic sort, 4096 keys, descending ----------------
__device__ void bitonic4096_desc(ull* sk) {
  for (unsigned k = 2; k <= 4096; k <<= 1) {
    for (unsigned j = k >> 1; j > 0; j >>= 1) {
      __syncthreads();
      for (unsigned i = threadIdx.x; i < 4096; i += 1024) {
        unsigned l = i ^ j;
        if (l > i) {
          ull a = sk[i], b = sk[l];
          bool desc = ((i & k) == 0);
          if ((a < b) == desc) { sk[i] = b; sk[l] = a; }
        }
      }
    }
  }
  __syncthreads();
}

// ---------------- sort candidates, emit per-level top-1000 outputs ----------------
__global__ __launch_bounds__(1024) void k_sort_cand(Ws* w, const float* __restrict__ reg,
                                                    const float* __restrict__ scales, int lvl,
                                                    float stride, int fw, int outBase, int total) {
  __shared__ ull sk[4096];
  unsigned cnt = w->candCount; if (cnt > CAND_CAP) cnt = CAND_CAP;
  for (unsigned i = threadIdx.x; i < 4096; i += 1024) sk[i] = (i < cnt) ? w->cand[i] : 0ull;
  bitonic4096_desc(sk);
  if (threadIdx.x < KTOP) {
    ull key = sk[threadIdx.x];
    unsigned bits = (unsigned)(key >> 32);
    unsigned idx  = ~(unsigned)(key & 0xFFFFFFFFull);
    if (idx >= (unsigned)total) idx = 0;               // safety (only hit if cand underflow)
    float score = __uint_as_float(bits);
    int a = (int)(idx / NCLS), lab = (int)(idx % NCLS);
    float s = scales[lvl];
    float r0 = fmaxf(reg[a * 4 + 0] * s, 0.0f) * stride;
    float r1 = fmaxf(reg[a * 4 + 1] * s, 0.0f) * stride;
    float r2 = fmaxf(reg[a * 4 + 2] * s, 0.0f) * stride;
    float r3 = fmaxf(reg[a * 4 + 3] * s, 0.0f) * stride;
    float ax = ((float)(a % fw) + 0.5f) * stride;
    float ay = ((float)(a / fw) + 0.5f) * stride;
    int o = outBase + (int)threadIdx.x;
    w->scores[o] = score;
    w->labels[o] = lab;
    w->boxes[o * 4 + 0] = ax - r0;
    w->boxes[o * 4 + 1] = ay - r1;
    w->boxes[o * 4 + 2] = ax + r2;
    w->boxes[o * 4 + 3] = ay + r3;
  }
}

// ---------------- global sort of 3000 detections (valid->score, invalid->-1) ----------------
__global__ __launch_bounds__(1024) void k_sort_global(Ws* w) {
  __shared__ ull sk[4096];
  for (unsigned i = threadIdx.x; i < 4096; i += 1024) {
    ull key = 0;
    if (i < NTOT) {
      float sc = w->scores[i];
      float eff = (sc > CONF) ? sc : -1.0f;
      unsigned b = __float_as_uint(eff);
      unsigned u = (b & 0x80000000u) ? ~b : (b | 0x80000000u);   // total order over floats
      key = ((ull)u << 32) | (unsigned)(~i);
    }
    sk[i] = key;
  }
  bitonic4096_desc(sk);
  for (unsigned i = threadIdx.x; i < NTOT; i += 1024)
    w->order[i] = (int)(~(unsigned)(sk[i] & 0xFFFFFFFFull));
}

// ---------------- suppression-mask bits (j > i, ovr > thresh); WMMA computes area_i+area_j ----------------
__device__ __forceinline__ unsigned jmask(int i, int jb) {
  unsigned m = 0xFFFFFFFFu;
  if (i >= jb) { int s = i - jb; m = (s >= 31) ? 0u : (0xFFFFFFFFu << (s + 1)); }
  int rem = NTOT - jb;
  if (rem <= 0) return 0u;
  if (rem < 32) m &= ((1u << rem) - 1u);
  return m;
}

__global__ __launch_bounds__(32) void k_iou(Ws* w) {
  int wt = blockIdx.x;           // word tile: cols [wt*32, wt*32+32)
  int rt = blockIdx.y;           // row tile:  rows [rt*16, rt*16+16)
  int lane = threadIdx.x;
  __shared__ float rx1[16], ry1[16], rx2[16], ry2[16], rar[16];
  if (lane < 16) {
    int r = rt * 16 + lane;
    int o = w->order[r < NTOT ? r : NTOT - 1];
    float off = (float)w->labels[o] * COFF;
    float x1 = w->boxes[o * 4 + 0] + off, y1 = w->boxes[o * 4 + 1] + off;
    float x2 = w->boxes[o * 4 + 2] + off, y2 = w->boxes[o * 4 + 3] + off;
    rx1[lane] = x1; ry1[lane] = y1; rx2[lane] = x2; ry2[lane] = y2;
    rar[lane] = (x2 - x1) * (y2 - y1);
  }
  __syncthreads();
  int hi = lane >> 4, ln = lane & 15;
  // A (16x4 f32): row M = ln holds [area_row, 1, 0, 0]; K=2,3 slots (lanes>=16) zero
  v2f A; A.x = (hi == 0) ? rar[ln] : 0.0f; A.y = (hi == 0) ? 1.0f : 0.0f;
  unsigned ball[2][8];
#pragma unroll
  for (int t = 0; t < 2; ++t) {
    int c = wt * 32 + t * 16 + ln;
    int oc = w->order[c < NTOT ? c : NTOT - 1];
    float off = (float)w->labels[oc] * COFF;
    float cx1 = w->boxes[oc * 4 + 0] + off, cy1 = w->boxes[oc * 4 + 1] + off;
    float cx2 = w->boxes[oc * 4 + 2] + off, cy2 = w->boxes[oc * 4 + 3] + off;
    float car = (cx2 - cx1) * (cy2 - cy1);
    // B (4x16 f32): col N = ln holds [1; area_col; 0; 0]
    v2f B; B.x = (hi == 0) ? 1.0f : 0.0f; B.y = (hi == 0) ? car : 0.0f;
    v8f zero = {};
    // sums[v] = area_row(v + 8*hi) + area_col(ln)  (C/D 16x16 f32 layout)
    v8f sums = __builtin_amdgcn_wmma_f32_16x16x4_f32(false, A, false, B, (short)0, zero,
                                                     false, false);
#pragma unroll
    for (int v = 0; v < 8; ++v) {
      int r = v + 8 * hi;
      float xx1 = fmaxf(rx1[r], cx1), yy1 = fmaxf(ry1[r], cy1);
      float xx2 = fminf(rx2[r], cx2), yy2 = fminf(ry2[r], cy2);
      float iw = fmaxf(1e-10f, xx2 - xx1), ih = fmaxf(1e-10f, yy2 - yy1);
      float inter = iw * ih;
      float ovr = inter / (sums[v] - inter + 1e-14f);
      ball[t][v] = (unsigned)__ballot(ovr > NMS_T);
    }
  }
  if (lane == 0) {
    int jb = wt * 32;
#pragma unroll
    for (int v = 0; v < 8; ++v) {
      int i0 = rt * 16 + v, i1 = i0 + 8;
      unsigned w0 = (ball[0][v] & 0xFFFFu) | ((ball[1][v] & 0xFFFFu) << 16);
      unsigned w1 = (ball[0][v] >> 16) | (ball[1][v] & 0xFFFF0000u);
      if (i0 < NTOT) w->maskmat[i0 * NWORDS + wt] = w0 & jmask(i0, jb);
      if (i1 < NTOT) w->maskmat[i1 * NWORDS + wt] = w1 & jmask(i1, jb);
    }
  }
}

// ---------------- sequential greedy NMS over bitmask + scatter keep ----------------
__global__ __launch_bounds__(128) void k_nms(Ws* w, float* out) {
  __shared__ unsigned keep[NWORDS];
  __shared__ int cur;
  int tid = threadIdx.x;
  if (tid < NWORDS) keep[tid] = 0;
  __syncthreads();
  for (int p = tid; p < NTOT; p += 128) {
    int o = w->order[p];
    if (w->scores[o] > CONF) atomicOr(&keep[p >> 5], 1u << (p & 31));
  }
  __syncthreads();
  for (int i = 0; i < NTOT; ++i) {
    if (tid == 0) cur = (int)((keep[i >> 5] >> (i & 31)) & 1u);
    __syncthreads();
    if (cur && tid < NWORDS) keep[tid] &= ~w->maskmat[i * NWORDS + tid];
    __syncthreads();
  }
  for (int p = tid; p < NTOT; p += 128) {
    int o = w->order[p];
    out[18000 + o] = ((keep[p >> 5] >> (p & 31)) & 1u) ? 1.0f : 0.0f;
  }
}

// ---------------- final outputs: normalized boxes, scores, labels ----------------
__global__ __launch_bounds__(256) void k_finalize(const Ws* w, float* out) {
  int t = blockIdx.x * 256 + threadIdx.x;
  if (t < 12000) {
    float v = w->boxes[t] * (1.0f / 4096.0f);
    out[t] = fminf(1.0f, fmaxf(0.0f, v));
  } else if (t < 15000) {
    out[t] = w->scores[t - 12000];
  } else if (t < 18000) {
    out[t] = (float)w->labels[t - 15000];
  }
}

extern "C" void kernel_launch(void* const* d_in, const int* in_sizes, int n_in,
                              void* d_out, int out_size, void* d_ws, size_t ws_size,
                              hipStream_t stream) {
  (void)in_sizes; (void)n_in; (void)out_size; (void)ws_size;
  Ws* w = (Ws*)d_ws;
  const float* scales = (const float*)d_in[9];
  float* out = (float*)d_out;
  static const int   totals[3]  = {512 * 512 * 80, 256 * 256 * 80, 128 * 128 * 80};
  static const int   fws[3]     = {512, 256, 128};
  static const float strides[3] = {8.0f, 16.0f, 32.0f};

  for (int lvl = 0; lvl < 3; ++lvl) {
    const float* cls = (const float*)d_in[3 * lvl + 0];
    const float* reg = (const float*)d_in[3 * lvl + 1];
    const float* ctn = (const float*)d_in[3 * lvl + 2];
    int total = totals[lvl];
    int nscan = total / 1024;                 // 256 threads x 4 elems
    k_zero<<<17, 256, 0, stream>>>(w);
    k_score_hist<<<nscan, 256, 0, stream>>>(cls, ctn, w);
    k_pick1<<<1, 32, 0, stream>>>(w);
    k_hist2<<<nscan, 256, 0, stream>>>(cls, ctn, w);
    k_pick2<<<1, 32, 0, stream>>>(w);
    k_compact<<<nscan, 256, 0, stream>>>(cls, ctn, w);
    k_sort_cand<<<1, 1024, 0, stream>>>(w, reg, scales, lvl, strides[lvl], fws[lvl],
                                        1000 * lvl, total);
  }
  k_sort_global<<<1, 1024, 0, stream>>>(w);
  k_iou<<<dim3(NWORDS, 188), 32, 0, stream>>>(w);
  k_nms<<<1, 128, 0, stream>>>(w, out);
  k_finalize<<<(18000 + 255) / 256, 256, 0, stream>>>(w, out);
}